// FCA_257698038332
// MI455X (gfx1250) — compile-verified
//
#include <hip/hip_runtime.h>
#include <math.h>

typedef float v8f __attribute__((ext_vector_type(8)));
typedef float v2f __attribute__((ext_vector_type(2)));

#define LD 68            // padded LDS row stride (floats): 16B-aligned rows, conflict-free transpose reads
#define NPLANE 4096      // B*C = 16*256
#define PLANE_ELEMS 4096 // 64*64

// ---------------------------------------------------------------------------
// Kernel 0: build the DCT-II matrix M[k][n] = cos(pi*(n+0.5)*k/64)/64
// ---------------------------------------------------------------------------
__global__ void fca_init_dctmat(float* __restrict__ Mmat) {
    int idx = blockIdx.x * 256 + threadIdx.x;       // 16 blocks * 256 = 4096
    int k = idx >> 6;
    int n = idx & 63;
    const float pi = 3.14159265358979323846f;
    Mmat[idx] = __cosf(pi * (n + 0.5f) * (float)k * (1.0f / 64.0f)) * (1.0f / 64.0f);
}

// ---------------------------------------------------------------------------
// Kernel 1: per plane, D = M * (X * M^T) via V_WMMA_F32_16X16X4_F32, then max.
// One 256-thread block (8 waves) per plane; each wave owns 2 of the 16 tiles.
// ---------------------------------------------------------------------------
__global__ void fca_dct_max(const float* __restrict__ ip,
                            const float* __restrict__ Mmat,
                            float* __restrict__ top1) {
    __shared__ __align__(16) float sX[64 * LD];
    __shared__ __align__(16) float sM[64 * LD];
    __shared__ __align__(16) float sT[64 * LD];
    __shared__ float sRed[8];

    const int t = threadIdx.x;
    const int plane = blockIdx.x;
    const float* __restrict__ src = ip + (size_t)plane * PLANE_ELEMS;

    // Stage X and M into LDS (1024 float4 each; 4 per thread)
    for (int i = t; i < 1024; i += 256) {
        int row = i >> 4;
        int c4  = (i & 15) << 2;
        float4 vx = *(const float4*)(src  + row * 64 + c4);
        float4 vm = *(const float4*)(Mmat + row * 64 + c4);
        *(float4*)&sX[row * LD + c4] = vx;
        *(float4*)&sM[row * LD + c4] = vm;
    }
    __syncthreads();

    const int wave = t >> 5;
    const int lane = t & 31;
    const int ll   = lane & 15;       // lane low
    const int hi   = lane >> 4;       // 0 or 1 (half-wave)
    const int k2   = hi << 1;         // K offset of this half-wave in A/B frags

    // ---------------- GEMM 1: T = X * M^T ----------------
    #pragma unroll
    for (int tt = 0; tt < 2; ++tt) {
        const int tile = wave + (tt << 3);            // tiles wave and wave+8
        const int m0 = (tile >> 2) << 4;
        const int n0 = (tile & 3) << 4;
        v8f acc = {};
        #pragma unroll
        for (int k0 = 0; k0 < 64; k0 += 4) {
            v2f a, b;
            // A = X[m0..m0+15][k0..k0+3]
            a.x = sX[(m0 + ll) * LD + k0 + k2];
            a.y = sX[(m0 + ll) * LD + k0 + k2 + 1];
            // B = M^T  =>  B[k][n] = M[n][k]  (row-major read of M)
            b.x = sM[(n0 + ll) * LD + k0 + k2];
            b.y = sM[(n0 + ll) * LD + k0 + k2 + 1];
            acc = __builtin_amdgcn_wmma_f32_16x16x4_f32(
                false, a, false, b, (short)0, acc, false, false);
        }
        // Scatter accumulator tile into sT
        #pragma unroll
        for (int i = 0; i < 8; ++i) {
            sT[(m0 + i + (hi << 3)) * LD + n0 + ll] = acc[i];
        }
    }
    __syncthreads();

    // ---------------- GEMM 2: D = M * T, fused max ----------------
    float mx = -INFINITY;
    #pragma unroll
    for (int tt = 0; tt < 2; ++tt) {
        const int tile = wave + (tt << 3);
        const int m0 = (tile >> 2) << 4;
        const int n0 = (tile & 3) << 4;
        v8f acc = {};
        #pragma unroll
        for (int k0 = 0; k0 < 64; k0 += 4) {
            v2f a, b;
            // A = M[m0..m0+15][k0..k0+3]
            a.x = sM[(m0 + ll) * LD + k0 + k2];
            a.y = sM[(m0 + ll) * LD + k0 + k2 + 1];
            // B = T[k0..k0+3][n0..n0+15] (row-major)
            b.x = sT[(k0 + k2) * LD + n0 + ll];
            b.y = sT[(k0 + k2 + 1) * LD + n0 + ll];
            acc = __builtin_amdgcn_wmma_f32_16x16x4_f32(
                false, a, false, b, (short)0, acc, false, false);
        }
        #pragma unroll
        for (int i = 0; i < 8; ++i) mx = fmaxf(mx, acc[i]);
    }

    // Wave32 butterfly max, then cross-wave via LDS
    #pragma unroll
    for (int off = 16; off >= 1; off >>= 1)
        mx = fmaxf(mx, __shfl_xor(mx, off, 32));
    if (lane == 0) sRed[wave] = mx;
    __syncthreads();
    if (t == 0) {
        float m = sRed[0];
        #pragma unroll
        for (int i = 1; i < 8; ++i) m = fmaxf(m, sRed[i]);
        top1[plane] = m;
    }
}

// ---------------------------------------------------------------------------
// Kernel 2: weights = sigmoid(relu(top1 @ w1) @ w2)   [tiny: one block]
// ---------------------------------------------------------------------------
__global__ void fca_mlp(const float* __restrict__ top1,
                        const float* __restrict__ w1,   // [256][16]
                        const float* __restrict__ w2,   // [16][256]
                        float* __restrict__ weights) {  // [16][256]
    __shared__ float sTop[16 * 256];
    __shared__ float sH[16 * 16];
    const int t = threadIdx.x;
    for (int i = t; i < 4096; i += 256) sTop[i] = top1[i];
    __syncthreads();

    // h[b][j] = relu(sum_c top1[b][c] * w1[c][j]);  t -> (b=t/16, j=t%16)
    {
        const int b = t >> 4, j = t & 15;
        float s = 0.0f;
        for (int c = 0; c < 256; ++c) s += sTop[b * 256 + c] * w1[c * 16 + j];
        sH[t] = fmaxf(s, 0.0f);
    }
    __syncthreads();

    // weights[b][c] = sigmoid(sum_j h[b][j] * w2[j][c])
    for (int i = t; i < 4096; i += 256) {
        const int b = i >> 8, c = i & 255;
        float s = 0.0f;
        #pragma unroll
        for (int j = 0; j < 16; ++j) s += sH[b * 16 + j] * w2[j * 256 + c];
        weights[i] = 1.0f / (1.0f + __expf(-s));
    }
}

// ---------------------------------------------------------------------------
// Kernel 3: out = ip * weights[plane]   (float4 streaming; ip re-read hits L2)
// ---------------------------------------------------------------------------
__global__ void fca_scale(const float* __restrict__ ip,
                          const float* __restrict__ weights,
                          float* __restrict__ out) {
    const float4* __restrict__ ip4 = (const float4*)ip;
    float4* __restrict__ out4 = (float4*)out;
    const int base = blockIdx.x * 1024 + threadIdx.x;
    #pragma unroll
    for (int j = 0; j < 4; ++j) {
        const int i4 = base + j * 256;          // 4096 blocks * 1024 = all 4.19M float4
        const float w = weights[i4 >> 10];      // plane = (i4*4)/4096
        float4 v = ip4[i4];
        v.x *= w; v.y *= w; v.z *= w; v.w *= w;
        out4[i4] = v;
    }
}

// ---------------------------------------------------------------------------
extern "C" void kernel_launch(void* const* d_in, const int* in_sizes, int n_in,
                              void* d_out, int out_size, void* d_ws, size_t ws_size,
                              hipStream_t stream) {
    const float* ip = (const float*)d_in[0];   // [16,256,64,64]
    const float* w1 = (const float*)d_in[1];   // [256,16]
    const float* w2 = (const float*)d_in[2];   // [16,256]
    float* out = (float*)d_out;

    float* wsf     = (float*)d_ws;
    float* Mmat    = wsf;            // 4096 floats
    float* top1    = wsf + 4096;     // 4096 floats
    float* weights = wsf + 8192;     // 4096 floats

    fca_init_dctmat<<<16, 256, 0, stream>>>(Mmat);
    fca_dct_max<<<NPLANE, 256, 0, stream>>>(ip, Mmat, top1);
    fca_mlp<<<1, 256, 0, stream>>>(top1, w1, w2, weights);
    fca_scale<<<NPLANE, 256, 0, stream>>>(ip, weights, out);
}